// TCAMSeeder_6932077215815
// MI455X (gfx1250) — compile-verified
//
#include <hip/hip_runtime.h>
#include <stdint.h>

// ---------------- problem constants (match reference) ----------------
#define B_      64
#define H_      512
#define W_      512
#define HW_     (H_*W_)
#define EPS_    1e-8f
#define NBG_    78643      // int(0.3 * 512*512)
#define NSEED_  100        // MIN_ == MAX_ == 100
#define ERODE_R 5          // (11-1)/2
#define ROWS_PB 8
#define CHUNK_  4096       // floats per TDM stage (16 KiB); 2 buffers = 32 KiB LDS

#if __has_builtin(__builtin_amdgcn_tensor_load_to_lds) && __has_builtin(__builtin_amdgcn_s_wait_tensorcnt)
#define HAVE_TDM 1
#else
#define HAVE_TDM 0
#endif

typedef uint32_t u32x4_t __attribute__((ext_vector_type(4)));
typedef int      i32x8_t __attribute__((ext_vector_type(8)));
typedef int      i32x4_t __attribute__((ext_vector_type(4)));

#if HAVE_TDM
// 1-D contiguous TDM transfer: n f32 elements global -> LDS (n <= 65535).
// D# per CDNA5 ISA: group0 = {count, lds_addr, global_addr, type=2},
// group1 = {data_size=4B, tensor_dim0=n, tensor_dim1=1, tile_dim0=n, tile_dim1=0, stride=n}.
__device__ __forceinline__ void tdm_load_1d(const void* gsrc, void* ldst, uint32_t n) {
  uint64_t ga  = (uint64_t)(uintptr_t)gsrc;
  uint32_t lds = (uint32_t)(uintptr_t)ldst;   // low 32 bits of flat shared ptr == LDS offset
  u32x4_t g0; i32x8_t g1; i32x4_t g2; i32x4_t g3; i32x8_t g4;
  g0[0] = 1u; g0[1] = lds; g0[2] = (uint32_t)ga;
  g0[3] = (uint32_t)((ga >> 32) & 0x01FFFFFFu) | (2u << 30);
  g1[0] = (int)(2u << 16);                      // data_size = 4 bytes
  g1[1] = (int)((n & 0xFFFFu) << 16);           // tensor_dim0[15:0]
  g1[2] = (int)((n >> 16) | (1u << 16));        // tensor_dim0 hi | tensor_dim1 = 1
  g1[3] = (int)((n & 0xFFFFu) << 16);           // tile_dim0 = n
  g1[4] = 0;                                    // tile_dim1 = 0 (1-D tile)
  g1[5] = (int)n; g1[6] = 0; g1[7] = 0;         // tensor_dim0_stride
  g2[0]=g2[1]=g2[2]=g2[3]=0;
  g3[0]=g3[1]=g3[2]=g3[3]=0;
  g4[0]=g4[1]=g4[2]=g4[3]=g4[4]=g4[5]=g4[6]=g4[7]=0;
  __builtin_amdgcn_tensor_load_to_lds(g0, g1, g2, g3, g4, 0);
}
#endif

// ---------------- workspace view ----------------
struct WS {
  float*    th;       // [B]
  int*      deg;      // [B]
  uint32_t* roi_sum;  // [B]
  int*      n_fg;     // [B]
  uint32_t* cut_fgv; int* tie_fgv;
  int* k_fgs; uint32_t* cut_fgs; int* tie_fgs;
  uint32_t* cut_bgv; int* tie_bgv;
  int* k_bgs; uint32_t* cut_bgs; int* tie_bgs;
  uint8_t* rowtmp; uint8_t* roi; uint8_t* fgseed; uint8_t* bgseed;
};

// ---------------- helpers ----------------
__device__ __forceinline__ uint32_t asc_map(float f) {
  uint32_t u = __float_as_uint(f);
  return (u & 0x80000000u) ? ~u : (u | 0x80000000u);
}
__device__ __forceinline__ float gumbelf(float u) { return -__logf(-__logf(u)); }

// MODE: 0 = fg value key (desc camr)  1 = bg value key (asc camb)
//       2 = fg score key (desc)       3 = bg score key (desc)
template<int MODE>
__device__ __forceinline__ uint32_t key_from(int i, float c, uint32_t r, float u,
                                             uint32_t mcut, int mtie) {
  if (MODE == 0) {
    float camr = c * (float)r + EPS_;
    return ~asc_map(camr);
  } else if (MODE == 1) {
    return asc_map(c + EPS_);
  } else if (MODE == 2) {
    float camr = c * (float)r + EPS_;
    uint32_t kv = ~asc_map(camr);
    float s = -__builtin_inff();
    if (kv < mcut || (kv == mcut && i <= mtie)) s = __logf(camr) + gumbelf(u);
    return ~asc_map(s);
  } else {
    float camb = c + EPS_;
    uint32_t kv = asc_map(camb);
    float s = -__builtin_inff();
    if (kv < mcut || (kv == mcut && i <= mtie)) {
      float pb = fmaxf(1.0f - camb, 0.0f) + EPS_;
      s = __logf(pb) + gumbelf(u);
    }
    return ~asc_map(s);
  }
}

// 4 consecutive elements: histogram accumulate for the value-radix passes
template<int MODE>
__device__ __forceinline__ void accum_val4(int i0, const float* xq, const uint8_t* rs,
                                           const float* us, uint32_t mcut, int mtie,
                                           int pass, uint32_t prefix, int shift,
                                           uint32_t* hist) {
  float4 xv = *(const float4*)xq;
  uchar4 r4; r4.x = r4.y = r4.z = r4.w = 0;
  if (MODE == 0 || MODE == 2) r4 = *(const uchar4*)(rs + i0);
  float4 u4; u4.x = u4.y = u4.z = u4.w = 0.0f;
  if (MODE >= 2) { __builtin_prefetch(us + i0 + CHUNK_, 0, 0); u4 = *(const float4*)(us + i0); }
  const float    xa[4] = {xv.x, xv.y, xv.z, xv.w};
  const uint32_t ra[4] = {r4.x, r4.y, r4.z, r4.w};
  const float    ua[4] = {u4.x, u4.y, u4.z, u4.w};
#pragma unroll
  for (int k = 0; k < 4; ++k) {
    uint32_t key = key_from<MODE>(i0 + k, xa[k], ra[k], ua[k], mcut, mtie);
    if (pass == 0 || (key >> (32 - 8 * pass)) == prefix)
      atomicAdd(&hist[(key >> shift) & 0xFFu], 1u);
  }
}

// 4 consecutive elements: histogram accumulate for the tie-index-radix passes
template<int MODE>
__device__ __forceinline__ void accum_tie4(int i0, const float* xq, const uint8_t* rs,
                                           const float* us, uint32_t mcut, int mtie,
                                           uint32_t cutoff, int pass, uint32_t ip, int shift,
                                           uint32_t* hist) {
  float4 xv = *(const float4*)xq;
  uchar4 r4; r4.x = r4.y = r4.z = r4.w = 0;
  if (MODE == 0 || MODE == 2) r4 = *(const uchar4*)(rs + i0);
  float4 u4; u4.x = u4.y = u4.z = u4.w = 0.0f;
  if (MODE >= 2) u4 = *(const float4*)(us + i0);
  const float    xa[4] = {xv.x, xv.y, xv.z, xv.w};
  const uint32_t ra[4] = {r4.x, r4.y, r4.z, r4.w};
  const float    ua[4] = {u4.x, u4.y, u4.z, u4.w};
#pragma unroll
  for (int k = 0; k < 4; ++k) {
    uint32_t key = key_from<MODE>(i0 + k, xa[k], ra[k], ua[k], mcut, mtie);
    if (key == cutoff) {
      uint32_t idx = (uint32_t)(i0 + k);
      if (pass == 0 || (idx >> (24 - 8 * pass)) == ip)
        atomicAdd(&hist[(idx >> shift) & 0xFFu], 1u);
    }
  }
}

// ---------------- per-sample min/max + threshold clamp ----------------
__global__ __launch_bounds__(1024) void stats_kernel(const float* __restrict__ x,
                                                     const float* __restrict__ rt, WS w) {
  const int b = blockIdx.x, tid = threadIdx.x;
  const float* xs = x + (size_t)b * HW_;
  float mn = __builtin_inff(), mx = -__builtin_inff();
  for (int i0 = tid * 4; i0 < HW_; i0 += 4096) {
    float4 v = *(const float4*)(xs + i0);
    mn = fminf(mn, fminf(fminf(v.x, v.y), fminf(v.z, v.w)));
    mx = fmaxf(mx, fmaxf(fmaxf(v.x, v.y), fmaxf(v.z, v.w)));
  }
  __shared__ float smn[1024], smx[1024];
  smn[tid] = mn; smx[tid] = mx; __syncthreads();
  for (int s = 512; s > 0; s >>= 1) {
    if (tid < s) { smn[tid] = fminf(smn[tid], smn[tid + s]); smx[tid] = fmaxf(smx[tid], smx[tid + s]); }
    __syncthreads();
  }
  if (tid == 0) {
    float t = rt[b];
    t = (t == 0.0f) ? 1.0f : ((t == 255.0f) ? 254.0f : t);
    w.th[b]  = t;
    w.deg[b] = (smn[0] == smx[0]) ? 1 : 0;
  }
}

// ---------------- row erosion, fed by the Tensor Data Mover ----------------
__global__ __launch_bounds__(256) void erode_rows_kernel(const float* __restrict__ x, WS w) {
  __shared__ float   tile[ROWS_PB * W_];
  __shared__ uint8_t binr[ROWS_PB * W_];
  const int b    = blockIdx.x;
  const int row0 = blockIdx.y * ROWS_PB;
  const int tid  = threadIdx.x;
  const float* src = x + (size_t)b * HW_ + (size_t)row0 * W_;

#if HAVE_TDM
  if (tid < 32) {   // one wave issues the DMA (wave-level, EXEC-independent)
    tdm_load_1d(src, &tile[0], ROWS_PB * W_);   // 8 contiguous rows = 4096 f32
    __builtin_amdgcn_s_wait_tensorcnt(0);
  }
  __syncthreads();
#else
  for (int j = tid; j < ROWS_PB * W_; j += 256) tile[j] = src[j];
  __syncthreads();
#endif

  const float t = w.th[b];
  for (int j = tid; j < ROWS_PB * W_; j += 256)
    binr[j] = (floorf(tile[j] * 255.0f) > t) ? (uint8_t)1 : (uint8_t)0;
  __syncthreads();

  uint8_t* dst = w.rowtmp + (size_t)b * HW_ + (size_t)row0 * W_;
  for (int j = tid; j < ROWS_PB * W_; j += 256) {
    const int xc = j & (W_ - 1);
    const int base = j - xc;
    const int lo = max(xc - ERODE_R, 0), hi = min(xc + ERODE_R, W_ - 1);
    uint8_t m = 1;
    for (int k = lo; k <= hi; ++k) m &= binr[base + k];
    dst[j] = m;
  }
}

// ---------------- column erosion (x4 vectorized) + block popcount of roi ----------------
__global__ __launch_bounds__(256) void erode_cols_kernel(WS w) {
  const int gid = blockIdx.x * 256 + threadIdx.x;   // over B*HW/4
  const int b = gid >> 16;                          // HW/4 = 2^16 quads per sample
  const int p = (gid & 0xFFFF) * 4;
  const int y = p >> 9, xc = p & (W_ - 1);
  const uint8_t* rt = w.rowtmp + (size_t)b * HW_;
  const int lo = max(y - ERODE_R, 0), hi = min(y + ERODE_R, H_ - 1);
  uchar4 m; m.x = m.y = m.z = m.w = 1;
  for (int yy = lo; yy <= hi; ++yy) {
    uchar4 v = *(const uchar4*)(rt + yy * W_ + xc);
    m.x &= v.x; m.y &= v.y; m.z &= v.z; m.w &= v.w;
  }
  *(uchar4*)(w.roi + (size_t)b * HW_ + p) = m;
  uint32_t cnt = (uint32_t)m.x + m.y + m.z + m.w;
  __shared__ uint32_t sc[256];
  sc[threadIdx.x] = cnt; __syncthreads();
  for (int s = 128; s > 0; s >>= 1) {
    if (threadIdx.x < s) sc[threadIdx.x] += sc[threadIdx.x + s];
    __syncthreads();
  }
  if (threadIdx.x == 0) atomicAdd(&w.roi_sum[b], sc[0]);
}

// ---------------- exact radix-select with TDM-staged x stream ----------------
// One 1024-thread block per sample. 4 value passes find the k-th smallest mapped key,
// 3 index passes reproduce the stable-sort tie-break exactly. The x stream (dominant
// operand) is double-buffered through LDS by the Tensor Data Mover: wave 0 issues
// chunk c+1 while all 32 waves histogram chunk c out of LDS.
template<int MODE>
__global__ __launch_bounds__(1024) void select_kernel(const float* __restrict__ x,
                                                      const float* __restrict__ ufg,
                                                      const float* __restrict__ ubg, WS w) {
#if HAVE_TDM
  __shared__ float xbuf[2][CHUNK_];
#endif
  __shared__ uint32_t hist[256];
  __shared__ uint32_t sh_p;
  __shared__ int      sh_k;

  const int b = blockIdx.x, tid = threadIdx.x;
  const float*   xs = x + (size_t)b * HW_;
  const uint8_t* rs = w.roi + (size_t)b * HW_;
  const float*   us = (MODE == 2) ? ufg + (size_t)b * HW_
                    : (MODE == 3) ? ubg + (size_t)b * HW_ : nullptr;

  uint32_t mcut = 0; int mtie = 0; int k;
  if      (MODE == 0) { k = (int)floorf(0.3f * (float)w.roi_sum[b]); }
  else if (MODE == 1) { k = NBG_; }
  else if (MODE == 2) { int nfg = w.n_fg[b]; k = nfg < NSEED_ ? nfg : NSEED_;
                        mcut = w.cut_fgv[b]; mtie = w.tie_fgv[b]; }
  else                { k = NSEED_; mcut = w.cut_bgv[b]; mtie = w.tie_bgv[b]; }

  if (MODE == 0 && tid == 0) w.n_fg[b]  = k;
  if (MODE == 2 && tid == 0) w.k_fgs[b] = k;
  if (MODE == 3 && tid == 0) w.k_bgs[b] = k;

  if (k <= 0) {   // block-uniform: safe early exit before any barrier
    if (tid == 0) {
      if      (MODE == 0) { w.cut_fgv[b] = 0u; w.tie_fgv[b] = -1; }
      else if (MODE == 2) { w.cut_fgs[b] = 0u; w.tie_fgs[b] = -1; }
    }
    return;
  }

  const int NCH = HW_ / CHUNK_;

  // ---- 4 passes over key bytes (MSB first) ----
  uint32_t prefix = 0; int kk = k;
  for (int pass = 0; pass < 4; ++pass) {
    if (tid < 256) hist[tid] = 0;
    const int shift = 24 - 8 * pass;
#if HAVE_TDM
    if (tid < 32) tdm_load_1d(xs, &xbuf[0][0], CHUNK_);
    for (int c = 0; c < NCH; ++c) {
      if (tid < 32) __builtin_amdgcn_s_wait_tensorcnt(0);
      __syncthreads();                                   // buf[c&1] valid (and hist zeroed)
      if (c + 1 < NCH && tid < 32)
        tdm_load_1d(xs + (size_t)(c + 1) * CHUNK_, &xbuf[(c + 1) & 1][0], CHUNK_);
      const int i0 = c * CHUNK_ + tid * 4;
      accum_val4<MODE>(i0, &xbuf[c & 1][tid * 4], rs, us, mcut, mtie, pass, prefix, shift, hist);
      __syncthreads();                                   // done reading buf[c&1]
    }
#else
    __syncthreads();
    for (int i0 = tid * 4; i0 < HW_; i0 += 4096)
      accum_val4<MODE>(i0, xs + i0, rs, us, mcut, mtie, pass, prefix, shift, hist);
    __syncthreads();
#endif
    if (tid == 0) {
      int cum = 0, bin = 0;
      for (; bin < 256; ++bin) { int h = (int)hist[bin]; if (cum + h >= kk) break; cum += h; }
      if (bin > 255) bin = 255;
      sh_p = (prefix << 8) | (uint32_t)bin; sh_k = kk - cum;
    }
    __syncthreads();
    prefix = sh_p; kk = sh_k;
    __syncthreads();
  }
  const uint32_t cutoff = prefix;

  // ---- stable tie-break: kk-th smallest index among keys == cutoff (18-bit index) ----
  uint32_t ip = 0; int rr = kk;
  for (int pass = 0; pass < 3; ++pass) {
    if (tid < 256) hist[tid] = 0;
    const int shift = 16 - 8 * pass;
#if HAVE_TDM
    if (tid < 32) tdm_load_1d(xs, &xbuf[0][0], CHUNK_);
    for (int c = 0; c < NCH; ++c) {
      if (tid < 32) __builtin_amdgcn_s_wait_tensorcnt(0);
      __syncthreads();
      if (c + 1 < NCH && tid < 32)
        tdm_load_1d(xs + (size_t)(c + 1) * CHUNK_, &xbuf[(c + 1) & 1][0], CHUNK_);
      const int i0 = c * CHUNK_ + tid * 4;
      accum_tie4<MODE>(i0, &xbuf[c & 1][tid * 4], rs, us, mcut, mtie, cutoff, pass, ip, shift, hist);
      __syncthreads();
    }
#else
    __syncthreads();
    for (int i0 = tid * 4; i0 < HW_; i0 += 4096)
      accum_tie4<MODE>(i0, xs + i0, rs, us, mcut, mtie, cutoff, pass, ip, shift, hist);
    __syncthreads();
#endif
    if (tid == 0) {
      int cum = 0, bin = 0;
      for (; bin < 256; ++bin) { int h = (int)hist[bin]; if (cum + h >= rr) break; cum += h; }
      if (bin > 255) bin = 255;
      sh_p = (ip << 8) | (uint32_t)bin; sh_k = rr - cum;
    }
    __syncthreads();
    ip = sh_p; rr = sh_k;
    __syncthreads();
  }

  if (tid == 0) {
    const int tie = (int)ip;
    if      (MODE == 0) { w.cut_fgv[b] = cutoff; w.tie_fgv[b] = tie; }
    else if (MODE == 1) { w.cut_bgv[b] = cutoff; w.tie_bgv[b] = tie; }
    else if (MODE == 2) { w.cut_fgs[b] = cutoff; w.tie_fgs[b] = tie; }
    else                { w.cut_bgs[b] = cutoff; w.tie_bgs[b] = tie; }
  }
}

// ---------------- emit seeds (selected == key < cut, or ==cut with idx <= tie) ----------------
template<int MODE> // 2: fg, 3: bg
__global__ __launch_bounds__(256) void emit_kernel(const float* __restrict__ x,
                                                   const float* __restrict__ ufg,
                                                   const float* __restrict__ ubg, WS w) {
  const int gid = blockIdx.x * 256 + threadIdx.x;
  const int b = gid >> 18; const int i = gid & (HW_ - 1);
  if (w.deg[b]) return;
  const int k = (MODE == 2) ? w.k_fgs[b] : w.k_bgs[b];
  if (k <= 0) return;
  const uint32_t mcut = (MODE == 2) ? w.cut_fgv[b] : w.cut_bgv[b];
  const int      mtie = (MODE == 2) ? w.tie_fgv[b] : w.tie_bgv[b];
  const uint32_t scut = (MODE == 2) ? w.cut_fgs[b] : w.cut_bgs[b];
  const int      stie = (MODE == 2) ? w.tie_fgs[b] : w.tie_bgs[b];
  const float*   xs = x + (size_t)b * HW_;
  const uint8_t* rs = w.roi + (size_t)b * HW_;
  const float*   us = ((MODE == 2) ? ufg : ubg) + (size_t)b * HW_;
  const float    c = xs[i];
  const uint32_t r = (MODE == 2) ? (uint32_t)rs[i] : 0u;
  const float    u = us[i];
  const uint32_t key = key_from<MODE>(i, c, r, u, mcut, mtie);
  if (key < scut || (key == scut && i <= stie)) {
    uint8_t* seed = (MODE == 2) ? w.fgseed : w.bgseed;
    seed[(size_t)b * HW_ + i] = 1;
  }
}

// ---------------- fused 3x3 dilation + mutual exclusion + label write ----------------
__global__ __launch_bounds__(256) void finalize_kernel(float* __restrict__ out, WS w) {
  const int gid = blockIdx.x * 256 + threadIdx.x;
  const int b = gid >> 18; const int p = gid & (HW_ - 1);
  const int y = p >> 9, xc = p & (W_ - 1);
  const uint8_t* fs = w.fgseed + (size_t)b * HW_;
  const uint8_t* bs = w.bgseed + (size_t)b * HW_;
  const int ly = max(y - 1, 0), hy = min(y + 1, H_ - 1);
  const int lx = max(xc - 1, 0), hx = min(xc + 1, W_ - 1);
  int fg = 0, bg = 0;
  for (int yy = ly; yy <= hy; ++yy)
    for (int xx = lx; xx <= hx; ++xx) { fg |= fs[yy * W_ + xx]; bg |= bs[yy * W_ + xx]; }
  if (fg && bg) { fg = 0; bg = 0; }
  out[gid] = fg ? 1.0f : (bg ? 0.0f : -255.0f);
}

// ---------------- host entry ----------------
extern "C" void kernel_launch(void* const* d_in, const int* in_sizes, int n_in,
                              void* d_out, int out_size, void* d_ws, size_t ws_size,
                              hipStream_t stream) {
  const float* x   = (const float*)d_in[0];   // (B,1,H,W)
  const float* rt  = (const float*)d_in[1];   // (B,)
  const float* ufg = (const float*)d_in[2];   // (B,HW)
  const float* ubg = (const float*)d_in[3];   // (B,HW)
  float* out = (float*)d_out;                 // (B,H,W) labels as float
  char* ws = (char*)d_ws;

  WS w;
  w.th      = (float*)   (ws + 0);
  w.deg     = (int*)     (ws + 256);
  w.roi_sum = (uint32_t*)(ws + 512);
  w.n_fg    = (int*)     (ws + 768);
  w.cut_fgv = (uint32_t*)(ws + 1024);
  w.tie_fgv = (int*)     (ws + 1280);
  w.k_fgs   = (int*)     (ws + 1536);
  w.cut_fgs = (uint32_t*)(ws + 1792);
  w.tie_fgs = (int*)     (ws + 2048);
  w.cut_bgv = (uint32_t*)(ws + 2304);
  w.tie_bgv = (int*)     (ws + 2560);
  w.k_bgs   = (int*)     (ws + 2816);
  w.cut_bgs = (uint32_t*)(ws + 3072);
  w.tie_bgs = (int*)     (ws + 3328);
  const size_t big = (size_t)B_ * HW_;        // 16 MiB per byte-plane
  w.rowtmp = (uint8_t*)(ws + 4096);
  w.roi    = w.rowtmp + big;
  w.fgseed = w.roi + big;
  w.bgseed = w.fgseed + big;

  // zero state we accumulate into (graph-capture-safe)
  (void)hipMemsetAsync(ws, 0, 4096, stream);
  (void)hipMemsetAsync(w.fgseed, 0, 2 * big, stream);   // fg+bg seeds are contiguous

  const int nall  = (B_ * HW_) / 256;         // elementwise kernels
  const int nquad = (B_ * HW_) / (4 * 256);   // x4-vectorized column erosion

  stats_kernel      <<<B_, 1024, 0, stream>>>(x, rt, w);
  erode_rows_kernel <<<dim3(B_, H_ / ROWS_PB), 256, 0, stream>>>(x, w);
  erode_cols_kernel <<<nquad, 256, 0, stream>>>(w);

  select_kernel<0>  <<<B_, 1024, 0, stream>>>(x, ufg, ubg, w);  // fg value cutoff (rank<n_fg mask)
  select_kernel<2>  <<<B_, 1024, 0, stream>>>(x, ufg, ubg, w);  // fg Gumbel score top-k
  emit_kernel<2>    <<<nall, 256, 0, stream>>>(x, ufg, ubg, w);

  select_kernel<1>  <<<B_, 1024, 0, stream>>>(x, ufg, ubg, w);  // bg value cutoff (rank<N_BG mask)
  select_kernel<3>  <<<B_, 1024, 0, stream>>>(x, ufg, ubg, w);  // bg Gumbel score top-k
  emit_kernel<3>    <<<nall, 256, 0, stream>>>(x, ufg, ubg, w);

  finalize_kernel   <<<nall, 256, 0, stream>>>(out, w);
}